// RoIAlignMatrixNet_13013750907325
// MI455X (gfx1250) — compile-verified
//
#include <hip/hip_runtime.h>
#include <hip/hip_bf16.h>

// RoIAlign (MatrixNet-style, 4-level FPN) for MI455X / gfx1250.
// Per RoI:  OUT[256ch x 36] = V[256ch x 196samples] x M[196 x 36]
//   V = bilinear samples (gathered from level-selected feature map)
//   M = constant 1/16 pooling matrix (14x14 samples -> 6x6 outputs)
// GEMM via V_WMMA_F32_16X16X4_F32 (f32 exact), K = 196 = 49*4.
// All channel-invariant work (tap offsets/weights) and the globally-constant
// B fragments are staged in LDS once per block.

#define SD      14          // samples per spatial dim (PH*SR)
#define NS      196         // SD*SD
#define NSTEP   49          // NS/4 K-steps
#define NOUT    36          // 6x6 pooled outputs per channel
#define CCH     256         // channels
#define OUTSTR  (CCH * NOUT)
#define NBTAB   (NSTEP * 3 * 32)   // B-fragment table entries (v2f each)

typedef __attribute__((ext_vector_type(2))) float v2f;
typedef __attribute__((ext_vector_type(8))) float v8f;

// constant pooling weight M[s][n]: 1/16 iff sample (sy,sx) inside 4x4 block of (oy,ox)
__device__ __forceinline__ float pool_w(int s, int oy, int ox, bool valid) {
    int sy = s / SD;
    int sx = s - sy * SD;
    unsigned dy = (unsigned)(sy - 2 * oy);
    unsigned dx = (unsigned)(sx - 2 * ox);
    return (valid & (dy < 4u) & (dx < 4u)) ? 0.0625f : 0.0f;
}

// 1D sample geometry: clamped taps + validity-folded lerp weights
__device__ __forceinline__ void geom1d(int s, float lo, float step, int H, int stride,
                                       int& t0, int& t1, float& w0, float& w1) {
    float off = (float)(s >> 1) + 0.25f + 0.5f * (float)(s & 1);
    float p   = lo + off * step;
    bool  v   = (p >= -1.0f) && (p <= (float)H);
    float pc  = fminf(fmaxf(p, 0.0f), (float)(H - 1));
    float p0f = floorf(pc);
    float lp  = pc - p0f;
    int   p0  = (int)p0f;
    int   p1  = min(p0 + 1, H - 1);
    t0 = p0 * stride;
    t1 = p1 * stride;
    w0 = v ? (1.0f - lp) : 0.0f;
    w1 = v ? lp : 0.0f;
}

__global__ __launch_bounds__(256) void roialign_wmma_kernel(
    const float* __restrict__ f0, const float* __restrict__ f1,
    const float* __restrict__ f2, const float* __restrict__ f3,
    const float* __restrict__ rois, float* __restrict__ out, int nroi_per_b) {

    __shared__ v2f    sBt[NBTAB];   // B fragments: [step][tile][lane] -> {bK, bK+1}
    __shared__ int4   sOff[NS];     // per-sample 4 tap offsets (elements)
    __shared__ float4 sW[NS];       // per-sample 4 bilinear weights (validity folded)

    const int roiIdx = blockIdx.x;
    const int b = roiIdx / nroi_per_b;
    const float* r = rois + (size_t)roiIdx * 7;

    // --- block-uniform RoI setup ---
    float x1 = r[1], y1 = r[2], x2 = r[3], y2 = r[4];
    int lev = (int)r[6];
    const float* fbase;
    int H;
    if (lev == 0)      { fbase = f0; H = 160; }
    else if (lev == 1) { fbase = f1; H = 80;  }
    else if (lev == 2) { fbase = f2; H = 40;  }
    else               { fbase = f3; H = 20;  }
    const int W = H;
    const float sc = (float)W * 0.05f;              // 8*W/160
    x1 *= sc; y1 *= sc; x2 *= sc; y2 *= sc;
    const float bw = fmaxf(x2 - x1, 1.0f) * (1.0f / 7.0f);
    const float bh = fmaxf(y2 - y1, 1.0f) * (1.0f / 7.0f);
    const float* __restrict__ fptr = fbase + (size_t)b * CCH * H * W;

    const int tid = threadIdx.x;

    // --- fill constant B-fragment table (RoI-independent) ---
    for (int idx = tid; idx < NBTAB; idx += 256) {
        int lane2 = idx & 31;
        int rest  = idx >> 5;            // step*3 + t
        int step  = rest / 3;
        int t     = rest - 3 * step;
        int m2    = lane2 & 15;
        int half2 = lane2 >> 4;
        int n     = t * 16 + m2;
        bool vld  = n < NOUT;
        int oy    = n / 6;
        int ox    = n - 6 * oy;
        int sA    = step * 4 + 2 * half2;
        v2f bb;
        bb.x = pool_w(sA,     oy, ox, vld);
        bb.y = pool_w(sA + 1, oy, ox, vld);
        sBt[idx] = bb;
    }

    // --- fill per-sample tap table (RoI geometry) ---
    if (tid < NS) {
        int sy = tid / SD;
        int sx = tid - sy * SD;
        int y0, y1o; float wy0, wy1;
        int x0, x1o; float wx0, wx1;
        geom1d(sy, y1, bh, H, W, y0, y1o, wy0, wy1);
        geom1d(sx, x1, bw, H, 1, x0, x1o, wx0, wx1);
        sOff[tid] = make_int4(y0 + x0, y0 + x1o, y1o + x0, y1o + x1o);
        sW[tid]   = make_float4(wy0 * wx0, wy0 * wx1, wy1 * wx0, wy1 * wx1);
    }
    __syncthreads();

    // --- WMMA GEMM loop ---
    const int lane = tid & 31;
    const int wave = tid >> 5;
    const int m    = lane & 15;       // A: channel row / B,D: output column
    const int half = lane >> 4;
    const v2f* __restrict__ bp = sBt + lane;

    float* orow = out + (size_t)roiIdx * OUTSTR;

    for (int cc = wave; cc < 16; cc += 8) {          // 2 channel-chunks per wave
        const int chHW = (cc * 16 + m) * H * W;      // per-lane channel base (elems)
        v8f acc0 = {}, acc1 = {}, acc2 = {};

        for (int step = 0; step < NSTEP; ++step) {
            const int sA = step * 4 + 2 * half;      // this lane's two K slots
            const int sB = sA + 1;

            int4   oA = sOff[sA];  float4 wA = sW[sA];
            int4   oB = sOff[sB];  float4 wB = sW[sB];

            v2f a;
            a.x = wA.x * fptr[chHW + oA.x] + wA.y * fptr[chHW + oA.y]
                + wA.z * fptr[chHW + oA.z] + wA.w * fptr[chHW + oA.w];
            a.y = wB.x * fptr[chHW + oB.x] + wB.y * fptr[chHW + oB.y]
                + wB.z * fptr[chHW + oB.z] + wB.w * fptr[chHW + oB.w];

            const v2f b0 = bp[step * 96];
            const v2f b1 = bp[step * 96 + 32];
            const v2f b2 = bp[step * 96 + 64];

            acc0 = __builtin_amdgcn_wmma_f32_16x16x4_f32(false, a, false, b0,
                                                         (short)0, acc0, false, false);
            acc1 = __builtin_amdgcn_wmma_f32_16x16x4_f32(false, a, false, b1,
                                                         (short)0, acc1, false, false);
            acc2 = __builtin_amdgcn_wmma_f32_16x16x4_f32(false, a, false, b2,
                                                         (short)0, acc2, false, false);
        }

        // D layout: VGPR v -> channel row v + 8*half, column = m
        #pragma unroll
        for (int v = 0; v < 8; ++v) {
            int ch = cc * 16 + v + 8 * half;
            float* oc = orow + ch * NOUT;
            oc[m]      = acc0[v];                    // n = 0..15
            oc[16 + m] = acc1[v];                    // n = 16..31
            if (m < 4) oc[32 + m] = acc2[v];         // n = 32..35 (rest padding)
        }
    }
}

extern "C" void kernel_launch(void* const* d_in, const int* in_sizes, int n_in,
                              void* d_out, int out_size, void* d_ws, size_t ws_size,
                              hipStream_t stream) {
    const float* f0   = (const float*)d_in[0];
    const float* f1   = (const float*)d_in[1];
    const float* f2   = (const float*)d_in[2];
    const float* f3   = (const float*)d_in[3];
    const float* rois = (const float*)d_in[4];
    float* out = (float*)d_out;

    const int nroi_total = in_sizes[4] / 7;          // B*N = 1024
    const int nroi_per_b = nroi_total / 2;           // B = 2

    roialign_wmma_kernel<<<nroi_total, 256, 0, stream>>>(f0, f1, f2, f3, rois,
                                                         out, nroi_per_b);
}